// LayerCRF_12652973654334
// MI455X (gfx1250) — compile-verified
//
#include <hip/hip_runtime.h>
#include <hip/hip_bf16.h>

#define NEG_VAL  (-9999.0f)
#define PAD_IDX  0
#define SOS_IDX  47
#define S_DIM    48
#define T_DIM    512
#define B_DIM    1024

typedef __attribute__((ext_vector_type(2))) float v2f;
typedef __attribute__((ext_vector_type(8))) float v8f;

// ds_swizzle SWAPX16: group-of-32, xor_mask=0x10, or_mask=0, and_mask=0x1f
#define SWIZZLE_SWAPX16 0x401F
__device__ __forceinline__ float half_swap_f32(float x) {
    return __int_as_float(
        __builtin_amdgcn_ds_swizzle(__float_as_int(x), SWIZZLE_SWAPX16));
}

// ---------------------------------------------------------------------------
// Forward (denominator) kernel: scaled-exp-domain CRF forward recursion.
// Block = 16 batches; 3 waves, wave w owns state tile [16w, 16w+16).
// Per step: D(16x16) = A(16x48) x B(48x16) via 12 V_WMMA_F32_16X16X4_F32
// (three 4-deep interleaved chains), w = exp(emis) .* D, one-step-stale
// per-batch rescale, double-buffered LDS transpose, ONE barrier per step,
// emissions software-pipelined, wave-uniform fast path with loads written
// directly into the loop-carried B-operand registers.
// ---------------------------------------------------------------------------
__global__ __launch_bounds__(96) void crf_forward_kernel(
    const float* __restrict__ features,    // (B,T,S)
    const float* __restrict__ mask,        // (B,T)
    const float* __restrict__ transition,  // (S,S), transition[i*S+j], i=new j=old
    float* __restrict__ denom)             // (B,)
{
    // v stored as pairs: vbuf[ph][j>>1][b] = { v[2*(j>>1)], v[2*(j>>1)+1] }
    __shared__ float2 vbuf[2][S_DIM / 2][16];
    __shared__ float  maxbuf[2][3][16];

    const int lane = threadIdx.x & 31;
    const int wid  = threadIdx.x >> 5;     // 0..2 : state tile
    const int b    = lane & 15;            // batch within tile (N index)
    const int hi   = lane >> 4;            // lane half selects K/M offsets
    const int b0   = blockIdx.x * 16;      // batch tile base

    // --- A operand resident in VGPRs: exp(transition) rows for this tile ---
    // A-layout (32-bit 16x4): chunk c, vgpr p: m = lane%16, k = 4c + p + 2*hi
    float a[12][2];
    {
        const int row = 16 * wid + b;      // output state i
#pragma unroll
        for (int c = 0; c < 12; ++c)
#pragma unroll
            for (int p = 0; p < 2; ++p) {
                const int col = 4 * c + p + 2 * hi;   // old state j
                a[c][p] = __expf(transition[row * S_DIM + col]); // exp(-9999)->0
            }
    }

    // --- init phase 0: v_0 = e_SOS ; scale = 1 ---
    for (int idx = threadIdx.x; idx < (S_DIM / 2) * 16; idx += 96) {
        const int pi = idx >> 4, bb = idx & 15;
        float2 v; v.x = (2 * pi == SOS_IDX) ? 1.0f : 0.0f;
                  v.y = (2 * pi + 1 == SOS_IDX) ? 1.0f : 0.0f;
        vbuf[0][pi][bb] = v;
    }
    if (threadIdx.x < 48)
        maxbuf[0][threadIdx.x >> 4][threadIdx.x & 15] = 1.0f;

    // loop-invariant per-lane element offsets; time term stays scalar so the
    // compiler can use SGPR-base + VGPR-offset global loads
    const int foff = (b0 + b) * T_DIM * S_DIM + 16 * wid + 8 * hi;
    const int moff = (b0 + b) * T_DIM;

    float vb[12][2];                        // B operand (v_t), B-layout
#pragma unroll
    for (int c = 0; c < 12; ++c) { vb[c][0] = 0.0f; vb[c][1] = 0.0f; }
    float sel_prev = 1.0f;                  // mask select pending for v load
    float logm     = 0.0f;                  // per-batch running log scale
    int   ph       = 0;                     // LDS read phase

    // --- software pipeline: preload emissions/mask for t = 0 ---
    float4 L0 = *reinterpret_cast<const float4*>(features + foff);
    float4 L1 = *reinterpret_cast<const float4*>(features + foff + 4);
    float  mcur = mask[moff];

#pragma unroll 1
    for (int t = 0; t < T_DIM; ++t) {
        __syncthreads();                    // phase `ph` stores now visible

        // --- load v_t (B-layout pairs) directly into the operand registers;
        //     mask blend only on the (wave-uniform) slow path ---
        if (__all(sel_prev != 0.0f)) {
#pragma unroll
            for (int c = 0; c < 12; ++c) {
                const float2 nvv = vbuf[ph][2 * c + hi][b];
                vb[c][0] = nvv.x; vb[c][1] = nvv.y;
            }
        } else {
#pragma unroll
            for (int c = 0; c < 12; ++c) {
                const float2 nvv = vbuf[ph][2 * c + hi][b];
                vb[c][0] += sel_prev * (nvv.x - vb[c][0]);
                vb[c][1] += sel_prev * (nvv.y - vb[c][1]);
            }
        }

        // one-step-stale per-batch rescale factor (off WMMA path)
        float cm = fmaxf(fmaxf(maxbuf[ph][0][b], maxbuf[ph][1][b]),
                         fmaxf(maxbuf[ph][2][b], 1e-30f));
        const float inv = __builtin_amdgcn_rcpf(cm);

        // exps of current-step emissions (loaded last iteration) — these
        // interleave with WMMA issue instead of serializing after it
        float ex[8];
        ex[0] = __expf(L0.x);  ex[1] = __expf(L0.y);
        ex[2] = __expf(L0.z);  ex[3] = __expf(L0.w);
        ex[4] = __expf(L1.x);  ex[5] = __expf(L1.y);
        ex[6] = __expf(L1.z);  ex[7] = __expf(L1.w);

        // --- U = M @ v : three interleaved 4-deep chains of f32 WMMAs ---
        v8f acc0 = {}, acc1 = {}, acc2 = {};
#pragma unroll
        for (int c = 0; c < 4; ++c) {
            v2f av0 = { a[c][0],      a[c][1]      };
            v2f bv0 = { vb[c][0],     vb[c][1]     };
            v2f av1 = { a[c + 4][0],  a[c + 4][1]  };
            v2f bv1 = { vb[c + 4][0], vb[c + 4][1] };
            v2f av2 = { a[c + 8][0],  a[c + 8][1]  };
            v2f bv2 = { vb[c + 8][0], vb[c + 8][1] };
            acc0 = __builtin_amdgcn_wmma_f32_16x16x4_f32(
                false, av0, false, bv0, (short)0, acc0, false, false);
            acc1 = __builtin_amdgcn_wmma_f32_16x16x4_f32(
                false, av1, false, bv1, (short)0, acc1, false, false);
            acc2 = __builtin_amdgcn_wmma_f32_16x16x4_f32(
                false, av2, false, bv2, (short)0, acc2, false, false);
        }
        const v8f acc = (acc0 + acc1) + acc2;

        // --- issue next step's emission/mask loads (clamped, branchless) ---
        const int tn = (t + 1 < T_DIM) ? t + 1 : T_DIM - 1;
        const float4 N0 = *reinterpret_cast<const float4*>(features + tn * S_DIM + foff);
        const float4 N1 = *reinterpret_cast<const float4*>(features + tn * S_DIM + foff + 4);
        const float  mnext = mask[moff + tn];

        // --- w = exp(emis) .* U ---
        float w[8];
#pragma unroll
        for (int r = 0; r < 8; ++r) w[r] = ex[r] * acc[r];

        // --- per-batch max of w -> phase ph^1 (used next step) ---
        float pm = fmaxf(fmaxf(fmaxf(w[0], w[1]), fmaxf(w[2], w[3])),
                         fmaxf(fmaxf(w[4], w[5]), fmaxf(w[6], w[7])));
        pm = fmaxf(pm, half_swap_f32(pm));
        maxbuf[ph ^ 1][wid][b] = pm;        // both lane halves: same value

        // --- normalize, mask bookkeeping, store D-layout pairs to ph^1 ---
        const float sel_cur = (mcur != 0.0f) ? 1.0f : 0.0f;
        logm += sel_cur * __logf(cm);
#pragma unroll
        for (int r = 0; r < 8; r += 2) {
            float2 wv; wv.x = w[r] * inv; wv.y = w[r + 1] * inv;
            vbuf[ph ^ 1][8 * wid + 4 * hi + (r >> 1)][b] = wv;
        }
        sel_prev = sel_cur;
        L0 = N0; L1 = N1; mcur = mnext;
        ph ^= 1;
    }

    __syncthreads();
    // --- denom[b] = logm + log(sum_j v_final[j]) ---
    float s = 0.0f;
#pragma unroll
    for (int c = 0; c < 12; ++c) {
        const float2 nv = vbuf[ph][2 * c + hi][b];
        s += vb[c][0] + sel_prev * (nv.x - vb[c][0]);
        s += vb[c][1] + sel_prev * (nv.y - vb[c][1]);
    }
    s += half_swap_f32(s);
    if (wid == 0 && lane < 16)
        denom[b0 + lane] = logm + __logf(s);
}

// ---------------------------------------------------------------------------
// Numerator + final subtract: out[b] = denom[b] - sum_t (emis + trans) * mask
// ---------------------------------------------------------------------------
__global__ __launch_bounds__(256) void crf_numer_kernel(
    const float* __restrict__ features, const int* __restrict__ states,
    const float* __restrict__ mask, const float* __restrict__ transition,
    const float* __restrict__ denom, float* __restrict__ out)
{
    __shared__ float red[256];
    const int bidx = blockIdx.x;
    const int tid  = threadIdx.x;
    const int*   st = states   + (size_t)bidx * T_DIM;
    const float* f  = features + (size_t)bidx * T_DIM * S_DIM;
    const float* mk = mask     + (size_t)bidx * T_DIM;

    float s = 0.0f;
#pragma unroll
    for (int t = tid; t < T_DIM; t += 256) {
        const int cur  = st[t];
        const int prev = (t == 0) ? SOS_IDX : st[t - 1];
        const float v  = f[t * S_DIM + cur] + transition[cur * S_DIM + prev];
        s += v * mk[t];
    }
    red[tid] = s;
    __syncthreads();
#pragma unroll
    for (int off = 128; off > 0; off >>= 1) {
        if (tid < off) red[tid] += red[tid + off];
        __syncthreads();
    }
    if (tid == 0) out[bidx] = denom[bidx] - red[0];
}

extern "C" void kernel_launch(void* const* d_in, const int* in_sizes, int n_in,
                              void* d_out, int out_size, void* d_ws, size_t ws_size,
                              hipStream_t stream) {
    const float* features   = (const float*)d_in[0];   // (B,T,S) f32
    const int*   states     = (const int*)  d_in[1];   // (B,T)   i32
    const float* mask       = (const float*)d_in[2];   // (B,T)   f32
    const float* transition = (const float*)d_in[3];   // (S,S)   f32
    float* out   = (float*)d_out;                      // (B,)    f32
    float* denom = (float*)d_ws;                       // B floats of scratch

    crf_forward_kernel<<<B_DIM / 16, 96, 0, stream>>>(features, mask, transition, denom);
    crf_numer_kernel<<<B_DIM, 256, 0, stream>>>(features, states, mask, transition,
                                                denom, out);
}